// GCN_58076547776807
// MI455X (gfx1250) — compile-verified
//
#include <hip/hip_runtime.h>

typedef __attribute__((ext_vector_type(2))) float v2f;
typedef __attribute__((ext_vector_type(8))) float v8f;

// ---------------- degree / GCN normalization ----------------
__global__ void k_deg_init(float* __restrict__ deg, int n) {
  int i = blockIdx.x * blockDim.x + threadIdx.x;
  if (i < n) deg[i] = 1.0f;  // self-loop contributes 1 to every node's degree
}

__global__ void k_deg_count(const int* __restrict__ dst, float* __restrict__ deg, int E) {
  int e = blockIdx.x * blockDim.x + threadIdx.x;
  if (e < E)
    __hip_atomic_fetch_add(&deg[dst[e]], 1.0f, __ATOMIC_RELAXED, __HIP_MEMORY_SCOPE_AGENT);
}

__global__ void k_dinv(float* __restrict__ deg, int n) {
  int i = blockIdx.x * blockDim.x + threadIdx.x;
  if (i < n) {
    float d = deg[i];
    deg[i] = (d > 0.0f) ? rsqrtf(d) : 0.0f;  // deg >= 1 always, guard matches ref
  }
}

// ---------------- dense projection via V_WMMA_F32_16X16X4_F32 ----------------
// C[M x Nc] = act(A)[M x K] @ W[K x Nc], row-major. One 16x16 C-tile per wave32.
// Launch: grid.x = M/16 blocks, blockDim.x = (Nc/16)*32 threads.
// A-frag (16x4 f32): lane L holds A[L&15][2*(L>>4)] and A[L&15][2*(L>>4)+1].
// B-frag mirrors A transposed. C/D: acc[v] = C[16*blk + v + 8*(L>>4)][L&15].
template <bool RELU>
__global__ void k_gemm_wmma(const float* __restrict__ A, const float* __restrict__ W,
                            float* __restrict__ C, int K, int Nc) {
  const int lane = threadIdx.x & 31;
  const int wave = threadIdx.x >> 5;
  const int row  = blockIdx.x * 16 + (lane & 15);
  const int col  = wave * 16 + (lane & 15);
  const int kh   = (lane >> 4) << 1;  // 0 or 2: K sub-pair owned by this half-wave

  const float* arow = A + (size_t)row * K;
  v8f acc = {};
  for (int k = 0; k < K; k += 4) {
    float a0 = arow[k + kh];
    float a1 = arow[k + kh + 1];
    if (RELU) { a0 = fmaxf(a0, 0.0f); a1 = fmaxf(a1, 0.0f); }
    v2f a, b;
    a.x = a0;
    a.y = a1;
    b.x = W[(size_t)(k + kh) * Nc + col];
    b.y = W[(size_t)(k + kh + 1) * Nc + col];
    // 8 args: (neg_a, A, neg_b, B, c_mod, C, reuse_a, reuse_b)
    acc = __builtin_amdgcn_wmma_f32_16x16x4_f32(false, a, false, b, (short)0, acc,
                                                false, false);
  }
  const int rbase = blockIdx.x * 16 + ((lane >> 4) << 3);
#pragma unroll
  for (int v = 0; v < 8; ++v)
    C[(size_t)(rbase + v) * Nc + col] = acc[v];
}

// ---------------- accumulator init: bias + self-loop term ----------------
// acc[i][f] = bias[f] + dinv[i]^2 * H[i][f]          (F = 1<<FSHIFT)
template <int FSHIFT>
__global__ void k_init_acc(const float* __restrict__ H, const float* __restrict__ dinv,
                           const float* __restrict__ bias, float* __restrict__ acc,
                           int total) {
  int gid = blockIdx.x * blockDim.x + threadIdx.x;
  if (gid >= total) return;
  int i = gid >> FSHIFT;
  int f = gid & ((1 << FSHIFT) - 1);
  float di = dinv[i];
  acc[gid] = bias[f] + di * di * H[gid];
}

// ---------------- edge scatter: acc[dst] += norm * H[src] ----------------
// Feature dim F = 4 << LPE_SHIFT; each thread owns one float4 of one edge.
// Gathers are L2-resident (H fits in 192MB L2); adds are global_atomic_add_f32.
template <int LPE_SHIFT>
__global__ void k_edge_scatter(const int* __restrict__ src, const int* __restrict__ dst,
                               const float* __restrict__ dinv, const float* __restrict__ H,
                               float* __restrict__ acc, int E) {
  const int F = 4 << LPE_SHIFT;
  int gid = blockIdx.x * blockDim.x + threadIdx.x;
  int e = gid >> LPE_SHIFT;
  if (e >= E) return;
  int f4 = (gid & ((1 << LPE_SHIFT) - 1)) << 2;
  int s = src[e];
  int d = dst[e];
  float nrm = dinv[s] * dinv[d];
  float4 m = *(const float4*)(H + (size_t)s * F + f4);
  float* op = acc + (size_t)d * F + f4;
  __hip_atomic_fetch_add(op + 0, m.x * nrm, __ATOMIC_RELAXED, __HIP_MEMORY_SCOPE_AGENT);
  __hip_atomic_fetch_add(op + 1, m.y * nrm, __ATOMIC_RELAXED, __HIP_MEMORY_SCOPE_AGENT);
  __hip_atomic_fetch_add(op + 2, m.z * nrm, __ATOMIC_RELAXED, __HIP_MEMORY_SCOPE_AGENT);
  __hip_atomic_fetch_add(op + 3, m.w * nrm, __ATOMIC_RELAXED, __HIP_MEMORY_SCOPE_AGENT);
}

extern "C" void kernel_launch(void* const* d_in, const int* in_sizes, int n_in,
                              void* d_out, int out_size, void* d_ws, size_t ws_size,
                              hipStream_t stream) {
  const float* x  = (const float*)d_in[0];
  const int*   ei = (const int*)d_in[1];
  const float* W1 = (const float*)d_in[2];
  const float* b1 = (const float*)d_in[3];
  const float* W2 = (const float*)d_in[4];
  const float* b2 = (const float*)d_in[5];
  float* out = (float*)d_out;

  const int FIN = 256, FHID = 128, FOUT = 64;
  const int N = in_sizes[0] / FIN;  // 100000 (divisible by 16)
  const int E = in_sizes[1] / 2;    // 1600000

  const int* srcI = ei;      // edge_index[0]
  const int* dstI = ei + E;  // edge_index[1]

  // workspace layout (all offsets 16B-aligned for N=100000)
  float* ws   = (float*)d_ws;
  float* dinv = ws;                        // N
  float* xw1  = ws + N;                    // N*FHID
  float* agg1 = xw1 + (size_t)N * FHID;    // N*FHID
  float* h2   = xw1;                       // reuse: N*FOUT <= N*FHID

  const int T = 256;

  // GCN normalization
  k_deg_init<<<(N + T - 1) / T, T, 0, stream>>>(dinv, N);
  k_deg_count<<<(E + T - 1) / T, T, 0, stream>>>(dstI, dinv, E);
  k_dinv<<<(N + T - 1) / T, T, 0, stream>>>(dinv, N);

  // ---- layer 1 ----
  k_gemm_wmma<false><<<N / 16, (FHID / 16) * 32, 0, stream>>>(x, W1, xw1, FIN, FHID);
  k_init_acc<7><<<(N * FHID + T - 1) / T, T, 0, stream>>>(xw1, dinv, b1, agg1, N * FHID);
  k_edge_scatter<5><<<(E * 32 + T - 1) / T, T, 0, stream>>>(srcI, dstI, dinv, xw1, agg1, E);

  // ---- layer 2 (ReLU fused into A-load of the GEMM) ----
  k_gemm_wmma<true><<<N / 16, (FOUT / 16) * 32, 0, stream>>>(agg1, W2, h2, FHID, FOUT);
  k_init_acc<6><<<(N * FOUT + T - 1) / T, T, 0, stream>>>(h2, dinv, b2, out, N * FOUT);
  k_edge_scatter<4><<<(E * 16 + T - 1) / T, T, 0, stream>>>(srcI, dstI, dinv, h2, out, E);
}